// GCFSignal_56410100466019
// MI455X (gfx1250) — compile-verified
//
#include <hip/hip_runtime.h>

// ---------------------------------------------------------------------------
// LightGCN 2-layer propagation on MI455X (gfx1250).
// Strategy: build CSR by destination (atomic-free fp accumulation), gather
// SpMM with rows resident in the 192MB L2, TDM (tensor_load_to_lds) staging
// of per-block CSR edge chunks, global_prefetch for next gather row.
// ---------------------------------------------------------------------------

typedef unsigned int v4u __attribute__((ext_vector_type(4)));
typedef int          v8i __attribute__((ext_vector_type(8)));
typedef int          v4i __attribute__((ext_vector_type(4)));

#define EMB_D 64
#define NODES_PER_BLOCK 128
#define STAGE_CAP 4096   // int2 pairs staged in LDS (32 KB)

// ------------------------------ small kernels ------------------------------

__global__ void gcf_zero_kernel(int* __restrict__ p, int n) {
    int i = blockIdx.x * blockDim.x + threadIdx.x;
    if (i < n) p[i] = 0;
}

__global__ void gcf_hist_kernel(const int* __restrict__ dst,
                                int* __restrict__ cnt, int e) {
    int i = blockIdx.x * blockDim.x + threadIdx.x;
    if (i < e) atomicAdd(&cnt[dst[i]], 1);
}

// Per-block exclusive scan of 1024 elements (4/thread), emits block sums.
__global__ __launch_bounds__(256)
void gcf_scan_block_kernel(const int* __restrict__ cnt,
                           int* __restrict__ rp,
                           int* __restrict__ bsums, int n) {
    __shared__ int sd[256];
    const int t = threadIdx.x;
    const int base = blockIdx.x * 1024 + t * 4;
    int v0 = 0, v1 = 0, v2 = 0, v3 = 0;
    if (base + 0 < n) v0 = cnt[base + 0];
    if (base + 1 < n) v1 = cnt[base + 1];
    if (base + 2 < n) v2 = cnt[base + 2];
    if (base + 3 < n) v3 = cnt[base + 3];
    const int s = v0 + v1 + v2 + v3;
    sd[t] = s;
    __syncthreads();
    for (int off = 1; off < 256; off <<= 1) {
        int x = 0;
        if (t >= off) x = sd[t - off];
        __syncthreads();
        if (t >= off) sd[t] += x;
        __syncthreads();
    }
    const int excl = sd[t] - s;
    if (base + 0 < n) rp[base + 0] = excl;
    if (base + 1 < n) rp[base + 1] = excl + v0;
    if (base + 2 < n) rp[base + 2] = excl + v0 + v1;
    if (base + 3 < n) rp[base + 3] = excl + v0 + v1 + v2;
    if (t == 255) bsums[blockIdx.x] = sd[255];
}

// Single-block exclusive scan of up to 1024 partials, in place.
__global__ __launch_bounds__(1024)
void gcf_scan_partials_kernel(int* __restrict__ bsums, int nb) {
    __shared__ int sd[1024];
    const int t = threadIdx.x;
    const int s = (t < nb) ? bsums[t] : 0;
    sd[t] = s;
    __syncthreads();
    for (int off = 1; off < 1024; off <<= 1) {
        int x = 0;
        if (t >= off) x = sd[t - off];
        __syncthreads();
        if (t >= off) sd[t] += x;
        __syncthreads();
    }
    if (t < nb) bsums[t] = sd[t] - s;
}

__global__ void gcf_finalize_kernel(int* __restrict__ rp,
                                    int* __restrict__ cursor,
                                    const int* __restrict__ bsums,
                                    int n, int total) {
    int i = blockIdx.x * blockDim.x + threadIdx.x;
    if (i < n) {
        const int v = rp[i] + bsums[i >> 10];
        rp[i] = v;
        cursor[i] = v;
    } else if (i == n) {
        rp[n] = total;
    }
}

__global__ void gcf_fill_kernel(const int* __restrict__ src,
                                const int* __restrict__ dst,
                                const float* __restrict__ w,
                                int* __restrict__ cursor,
                                int2* __restrict__ pairs, int e) {
    int i = blockIdx.x * blockDim.x + threadIdx.x;
    if (i < e) {
        const int d = dst[i];
        const int pos = atomicAdd(&cursor[d], 1);
        pairs[pos] = make_int2(src[i], __float_as_int(w[i]));
    }
}

// ------------------------------ SpMM kernels -------------------------------

__device__ __forceinline__ const float* x0_row(const float* __restrict__ u,
                                               const float* __restrict__ it,
                                               int r, int nu) {
    return (r < nu) ? (u + (size_t)r * EMB_D)
                    : (it + (size_t)(r - nu) * EMB_D);
}

// LAYER 0: outp[d,:] = sum_e w * x0[src,:]            (outp = B1)
// LAYER 1: outp[d,:] = (x0[d,:] + b1[d,:] + sum_e w * b1[src,:]) / 3
template <int LAYER>
__global__ __launch_bounds__(256)
void gcf_spmm_kernel(const int* __restrict__ row_ptr,
                     const int2* __restrict__ pairs,
                     const float* __restrict__ u_embs,
                     const float* __restrict__ i_embs,
                     const float* __restrict__ b1,
                     float* __restrict__ outp,
                     int num_nodes, int num_users) {
    __shared__ int2 s_pairs[STAGE_CAP];

    const int base    = blockIdx.x * NODES_PER_BLOCK;
    const int lane16  = threadIdx.x & 15;
    const int group   = threadIdx.x >> 4;   // 16 groups of 16 lanes
    const int nodeEnd = min(base + NODES_PER_BLOCK, num_nodes);

    const int eBeg  = row_ptr[base];
    const int eEnd  = row_ptr[nodeEnd];
    const int chunk = eEnd - eBeg;
    const bool staged = (chunk > 0) && (chunk <= STAGE_CAP);

    if (staged) {
        if (threadIdx.x < 32) {  // wave 0 issues the TDM DMA for the block
            const unsigned sBeg   = (unsigned)__builtin_amdgcn_readfirstlane(eBeg);
            const unsigned sChunk = (unsigned)__builtin_amdgcn_readfirstlane(chunk);
            const unsigned long long ga =
                (unsigned long long)(const void*)(pairs + sBeg);
            const unsigned lds_addr =
                (unsigned)(unsigned long long)(const void*)&s_pairs[0];

            // D# group 0: count=1 | lds_addr | global_addr[56:0] | type=2
            v4u g0;
            g0[0] = 1u;
            g0[1] = lds_addr;
            g0[2] = (unsigned)ga;
            g0[3] = (unsigned)(ga >> 32) | (2u << 30);

            // D# group 1: data_size=8B pairs; degenerate 2D tile
            // tensor_dim0 = tile_dim0 = chunk, tensor_dim1 = tile_dim1 = 1
            v8i g1;
            g1[0] = (int)(3u << 16);                                // data_size=3 (8B)
            g1[1] = (int)((sChunk & 0xFFFFu) << 16);                // tensor_dim0 lo16
            g1[2] = (int)(((sChunk >> 16) & 0xFFFFu) | (1u << 16)); // td0 hi16 | td1 lo16
            g1[3] = (int)((sChunk & 0xFFFFu) << 16);                // td1 hi16=0 | tile_dim0
            g1[4] = 1;                                              // tile_dim1=1, tile_dim2=0
            g1[5] = (int)sChunk;                                    // tensor_dim0_stride lo32
            g1[6] = 0;
            g1[7] = 0;

            v4i z4; z4[0] = 0; z4[1] = 0; z4[2] = 0; z4[3] = 0;
            v8i z8;
            z8[0] = 0; z8[1] = 0; z8[2] = 0; z8[3] = 0;
            z8[4] = 0; z8[5] = 0; z8[6] = 0; z8[7] = 0;
            __builtin_amdgcn_tensor_load_to_lds(g0, g1, z4, z4, z8, 0);
            __builtin_amdgcn_s_wait_tensorcnt(0);
        }
    }
    __syncthreads();

    const float inv3 = 1.0f / 3.0f;

    for (int n = base + group; n < nodeEnd; n += 16) {
        const int rb = row_ptr[n];
        const int re = row_ptr[n + 1];
        float4 acc = make_float4(0.f, 0.f, 0.f, 0.f);

        for (int e = rb; e < re; ++e) {
            const int2 pr = staged ? s_pairs[e - eBeg] : pairs[e];

            // prefetch next edge's gather row into cache
            if (e + 1 < re) {
                const int2 nx = staged ? s_pairs[e + 1 - eBeg] : pairs[e + 1];
                const float* nrow = (LAYER == 0)
                    ? x0_row(u_embs, i_embs, nx.x, num_users)
                    : (b1 + (size_t)nx.x * EMB_D);
                __builtin_prefetch(nrow + lane16 * 4, 0, 1);
            }

            const float w = __int_as_float(pr.y);
            const float* row = (LAYER == 0)
                ? x0_row(u_embs, i_embs, pr.x, num_users)
                : (b1 + (size_t)pr.x * EMB_D);
            const float4 v = ((const float4*)row)[lane16];
            acc.x = fmaf(w, v.x, acc.x);
            acc.y = fmaf(w, v.y, acc.y);
            acc.z = fmaf(w, v.z, acc.z);
            acc.w = fmaf(w, v.w, acc.w);
        }

        if (LAYER == 0) {
            ((float4*)(outp + (size_t)n * EMB_D))[lane16] = acc;
        } else {
            const float* x0r = x0_row(u_embs, i_embs, n, num_users);
            const float4 a = ((const float4*)x0r)[lane16];
            const float4 b = ((const float4*)(b1 + (size_t)n * EMB_D))[lane16];
            float4 o;
            o.x = (a.x + b.x + acc.x) * inv3;
            o.y = (a.y + b.y + acc.y) * inv3;
            o.z = (a.z + b.z + acc.z) * inv3;
            o.w = (a.w + b.w + acc.w) * inv3;
            ((float4*)(outp + (size_t)n * EMB_D))[lane16] = o;
        }
    }
}

// ------------------------------ host launcher ------------------------------

static inline size_t gcf_align256(size_t x) { return (x + 255) & ~(size_t)255; }

extern "C" void kernel_launch(void* const* d_in, const int* in_sizes, int n_in,
                              void* d_out, int out_size, void* d_ws, size_t ws_size,
                              hipStream_t stream) {
    (void)n_in; (void)out_size; (void)ws_size;

    const float* u_embs = (const float*)d_in[0];
    const float* i_embs = (const float*)d_in[1];
    const int*   esrc   = (const int*)d_in[2];
    const int*   edst   = (const int*)d_in[3];
    const float* ew     = (const float*)d_in[4];

    const int nu = in_sizes[0] / EMB_D;
    const int ni = in_sizes[1] / EMB_D;
    const int N  = nu + ni;
    const int E  = in_sizes[2];

    // workspace layout
    char* ws = (char*)d_ws;
    size_t o = 0;
    int*  rp     = (int*)(ws + o);  o = gcf_align256(o + (size_t)(N + 1) * 4);
    int*  cursor = (int*)(ws + o);  o = gcf_align256(o + (size_t)N * 4);
    int*  cnt    = (int*)(ws + o);  o = gcf_align256(o + (size_t)N * 4);
    int*  bsums  = (int*)(ws + o);  o = gcf_align256(o + 1024 * 4);
    int2* pairs  = (int2*)(ws + o); o = gcf_align256(o + (size_t)E * 8);
    float* b1    = (float*)(ws + o);

    const int nb1024 = (N + 1023) / 1024;

    gcf_zero_kernel<<<(N + 255) / 256, 256, 0, stream>>>(cnt, N);
    gcf_hist_kernel<<<(E + 255) / 256, 256, 0, stream>>>(edst, cnt, E);
    gcf_scan_block_kernel<<<nb1024, 256, 0, stream>>>(cnt, rp, bsums, N);
    gcf_scan_partials_kernel<<<1, 1024, 0, stream>>>(bsums, nb1024);
    gcf_finalize_kernel<<<(N + 1 + 255) / 256, 256, 0, stream>>>(rp, cursor, bsums, N, E);
    gcf_fill_kernel<<<(E + 255) / 256, 256, 0, stream>>>(esrc, edst, ew, cursor, pairs, E);

    const int nblocks = (N + NODES_PER_BLOCK - 1) / NODES_PER_BLOCK;
    gcf_spmm_kernel<0><<<nblocks, 256, 0, stream>>>(rp, pairs, u_embs, i_embs,
                                                    b1, b1, N, nu);
    gcf_spmm_kernel<1><<<nblocks, 256, 0, stream>>>(rp, pairs, u_embs, i_embs,
                                                    b1, (float*)d_out, N, nu);
}